// LiquidNeuralNetwork_9131100471630
// MI455X (gfx1250) — compile-verified
//
#include <hip/hip_runtime.h>
#include <hip/hip_bf16.h>
#include <math.h>

// ---------------------------------------------------------------------------
// LiquidNeuralNetwork on MI455X (gfx1250), f32 WMMA path with LDS staging.
//   stem:  resid = x@Wr+br ; h = tanh(tanh(x@Wi1+bi1)@Wi2+bi2) + resid
//   5x ODE blocks integrated 0..1 with fixed-step dopri5 (32 steps, k1..k6)
//   head:  out = tanh(tanh(h@Wo1+bo1)@Wo2+bo2)
// GEMM: 128x128 workgroup tile, K-slab 16 double-buffered in LDS,
//       wave tile 64x32 (4x2 WMMA tiles), v_wmma_f32_16x16x4_f32.
// ---------------------------------------------------------------------------

typedef float v2f __attribute__((ext_vector_type(2)));
typedef float v8f __attribute__((ext_vector_type(8)));

#define BM 128
#define BN 128
#define BK 16
#define LDA 20          // padded A row stride (floats): 80B keeps b128 aligned,
                        // gcd(20,64)=4 -> conflict-free ds_load_b64 A frags
#define NSTEPS 32

#define BATCH 1024
#define DIN   256
#define H     512
#define H2    1024
#define DOUT  128

// C[M,N] = act(A[M,K] @ W[K,N] + bias[N]),  act: 0=none, 1=tanh
// 256 threads = 8 waves arranged 2(M) x 4(N); wave tile 64x32.
__global__ __launch_bounds__(256) void gemm_bias_act(
    const float* __restrict__ A, const float* __restrict__ W,
    const float* __restrict__ bias, float* __restrict__ C,
    int M, int N, int K, int act)
{
    __shared__ float As[2][BM * LDA];
    __shared__ float Bs[2][BK * BN];

    const int tid   = threadIdx.x;
    const int wave  = tid >> 5;
    const int lane  = tid & 31;
    const int l16   = lane & 15;    // M (A frags) / N (B,C frags) within 16-tile
    const int lhi   = lane >> 4;    // K-pair select (A/B), row-half (C)
    const int waveM = wave >> 2;    // 0..1
    const int waveN = wave & 3;     // 0..3

    const int mBlock = blockIdx.y * BM;
    const int nBlock = blockIdx.x * BN;

    // global->LDS staging assignment (per thread: 2 float4 of A, 2 float4 of B)
    const int arow = tid >> 2;          // 0..63  (A rows, +64 for second half)
    const int acol = (tid & 3) * 4;     // 0,4,8,12
    const int brow = tid >> 5;          // 0..7   (B rows, +8 for second half)
    const int bcol = (tid & 31) * 4;    // 0..124

    v8f acc[4][2];
    #pragma unroll
    for (int mt = 0; mt < 4; ++mt)
        #pragma unroll
        for (int nt = 0; nt < 2; ++nt) acc[mt][nt] = {};

    float4 aReg0, aReg1, bReg0, bReg1;

    const float* aBase = A + (size_t)(mBlock + arow) * K + acol;
    const float* wBase = W + (size_t)brow * N + nBlock + bcol;

    // ---- preload slab 0 ----
    {
        const float* ap = aBase;
        aReg0 = *(const float4*)ap;
        aReg1 = *(const float4*)(ap + (size_t)64 * K);
        const float* bp = wBase;
        bReg0 = *(const float4*)bp;
        bReg1 = *(const float4*)(bp + (size_t)8 * N);
        *(float4*)&As[0][arow * LDA + acol]        = aReg0;
        *(float4*)&As[0][(arow + 64) * LDA + acol] = aReg1;
        *(float4*)&Bs[0][brow * BN + bcol]         = bReg0;
        *(float4*)&Bs[0][(brow + 8) * BN + bcol]   = bReg1;
    }
    __syncthreads();

    const int nSlabs = K / BK;
    for (int s = 0; s < nSlabs; ++s) {
        const int cur = s & 1;
        const int nxt = cur ^ 1;

        // issue global loads for next slab early (hide under 32 WMMAs)
        if (s + 1 < nSlabs) {
            const int kk = (s + 1) * BK;
            const float* ap = aBase + kk;
            aReg0 = *(const float4*)ap;
            aReg1 = *(const float4*)(ap + (size_t)64 * K);
            const float* bp = wBase + (size_t)kk * N;
            bReg0 = *(const float4*)bp;
            bReg1 = *(const float4*)(bp + (size_t)8 * N);
            // warm WGP$ two slabs ahead on the weight stream
            __builtin_prefetch(bp + (size_t)BK * N, 0, 1);
        }

        // ---- compute: 4 k-steps of 4, 8 WMMAs each ----
        #pragma unroll
        for (int ks = 0; ks < 4; ++ks) {
            const int ka = ks * 4 + lhi * 2;   // this lane's K base
            v2f a[4], b[2];
            #pragma unroll
            for (int mt = 0; mt < 4; ++mt) {
                const float* p =
                    &As[cur][(waveM * 64 + mt * 16 + l16) * LDA + ka];
                a[mt].x = p[0];
                a[mt].y = p[1];
            }
            #pragma unroll
            for (int nt = 0; nt < 2; ++nt) {
                const int col = waveN * 32 + nt * 16 + l16;
                b[nt].x = Bs[cur][ka * BN + col];
                b[nt].y = Bs[cur][(ka + 1) * BN + col];
            }
            #pragma unroll
            for (int mt = 0; mt < 4; ++mt)
                #pragma unroll
                for (int nt = 0; nt < 2; ++nt)
                    acc[mt][nt] = __builtin_amdgcn_wmma_f32_16x16x4_f32(
                        false, a[mt], false, b[nt], (short)0, acc[mt][nt],
                        false, false);
        }

        // ---- stage next slab into the other buffer ----
        if (s + 1 < nSlabs) {
            *(float4*)&As[nxt][arow * LDA + acol]        = aReg0;
            *(float4*)&As[nxt][(arow + 64) * LDA + acol] = aReg1;
            *(float4*)&Bs[nxt][brow * BN + bcol]         = bReg0;
            *(float4*)&Bs[nxt][(brow + 8) * BN + bcol]   = bReg1;
        }
        __syncthreads();
    }

    // Epilogue: C layout — VGPR r holds row (r + 8*lhi), lane%16 = col.
    #pragma unroll
    for (int mt = 0; mt < 4; ++mt) {
        #pragma unroll
        for (int nt = 0; nt < 2; ++nt) {
            const int col = nBlock + waveN * 32 + nt * 16 + l16;
            const float bv = bias[col];
            float* cp = C + (size_t)(mBlock + waveM * 64 + mt * 16 + lhi * 8) * N
                          + col;
            v8f v = acc[mt][nt];
            #pragma unroll
            for (int r = 0; r < 8; ++r) {
                float val = v[r] + bv;
                if (act) val = tanhf(val);
                cp[(size_t)r * N] = val;
            }
        }
    }
}

// out[i] = base[i] + h * (c1*k1[i] + ... + c6*k6[i]); cj==0 => kj unused (may be null)
__global__ __launch_bounds__(256) void combine6(
    float* __restrict__ out, const float* __restrict__ base, float h,
    float c1, const float* __restrict__ k1,
    float c2, const float* __restrict__ k2,
    float c3, const float* __restrict__ k3,
    float c4, const float* __restrict__ k4,
    float c5, const float* __restrict__ k5,
    float c6, const float* __restrict__ k6, int n)
{
    const int i = blockIdx.x * blockDim.x + threadIdx.x;
    if (i >= n) return;
    float s = 0.0f;
    if (c1 != 0.0f) s += c1 * k1[i];
    if (c2 != 0.0f) s += c2 * k2[i];
    if (c3 != 0.0f) s += c3 * k3[i];
    if (c4 != 0.0f) s += c4 * k4[i];
    if (c5 != 0.0f) s += c5 * k5[i];
    if (c6 != 0.0f) s += c6 * k6[i];
    out[i] = base[i] + h * s;
}

// ---------------------------------------------------------------------------
// Host-side orchestration
// ---------------------------------------------------------------------------
static inline void launch_gemm(hipStream_t s, const float* A, const float* W,
                               const float* b, float* C, int M, int N, int K, int act)
{
    dim3 grid(N / BN, M / BM);
    gemm_bias_act<<<grid, 256, 0, s>>>(A, W, b, C, M, N, K, act);
}

// f(v) = tanh(tanh(v@W1+b1)@W2+b2)@W3+b3      (t unused: autonomous)
static inline void ode_f(hipStream_t s, const float* v, float* out,
                         const float* W1, const float* b1,
                         const float* W2, const float* b2,
                         const float* W3, const float* b3,
                         float* t1, float* t2)
{
    launch_gemm(s, v,  W1, b1, t1,  BATCH, H2, H,  1);
    launch_gemm(s, t1, W2, b2, t2,  BATCH, H2, H2, 1);
    launch_gemm(s, t2, W3, b3, out, BATCH, H,  H2, 0);
}

extern "C" void kernel_launch(void* const* d_in, const int* in_sizes, int n_in,
                              void* d_out, int out_size, void* d_ws, size_t ws_size,
                              hipStream_t stream)
{
    const float* x   = (const float*)d_in[0];
    const float* Wi1 = (const float*)d_in[1];
    const float* bi1 = (const float*)d_in[2];
    const float* Wi2 = (const float*)d_in[3];
    const float* bi2 = (const float*)d_in[4];
    const float* Wr  = (const float*)d_in[5];
    const float* br  = (const float*)d_in[6];
    const float* oW1 = (const float*)d_in[7];   // [5,512,1024]
    const float* ob1 = (const float*)d_in[8];   // [5,1024]
    const float* oW2 = (const float*)d_in[9];   // [5,1024,1024]
    const float* ob2 = (const float*)d_in[10];  // [5,1024]
    const float* oW3 = (const float*)d_in[11];  // [5,1024,512]
    const float* ob3 = (const float*)d_in[12];  // [5,512]
    const float* Wo1 = (const float*)d_in[13];
    const float* bo1 = (const float*)d_in[14];
    const float* Wo2 = (const float*)d_in[15];
    const float* bo2 = (const float*)d_in[16];

    const int nState = BATCH * H;               // 524288 floats
    float* ws   = (float*)d_ws;
    float* y    = ws;              ws += nState;
    float* ytmp = ws;              ws += nState;
    float* resid= ws;              ws += nState;
    float* k[6];
    for (int i = 0; i < 6; ++i) { k[i] = ws; ws += nState; }
    float* t1   = ws;              ws += BATCH * H2;
    float* t2   = ws;              ws += BATCH * H2;

    const int cb = (nState + 255) / 256;

    // ---- input stem ----
    launch_gemm(stream, x,  Wr,  br,  resid, BATCH, H, DIN, 0);
    launch_gemm(stream, x,  Wi1, bi1, t1,    BATCH, H, DIN, 1);
    launch_gemm(stream, t1, Wi2, bi2, ytmp,  BATCH, H, H,   1);
    combine6<<<cb, 256, 0, stream>>>(y, ytmp, 1.0f,
        1.0f, resid, 0.f, nullptr, 0.f, nullptr, 0.f, nullptr, 0.f, nullptr, 0.f, nullptr,
        nState);

    // ---- dopri5 tableau (fixed step) ----
    const float a21 = 0.2f;
    const float a31 = 3.0f/40.0f,      a32 = 9.0f/40.0f;
    const float a41 = 44.0f/45.0f,     a42 = -56.0f/15.0f,    a43 = 32.0f/9.0f;
    const float a51 = 19372.0f/6561.0f,a52 = -25360.0f/2187.0f,
                a53 = 64448.0f/6561.0f,a54 = -212.0f/729.0f;
    const float a61 = 9017.0f/3168.0f, a62 = -355.0f/33.0f,
                a63 = 46732.0f/5247.0f,a64 = 49.0f/176.0f,    a65 = -5103.0f/18656.0f;
    const float b1 = 35.0f/384.0f, b3 = 500.0f/1113.0f, b4 = 125.0f/192.0f,
                b5 = -2187.0f/6784.0f, b6 = 11.0f/84.0f;
    const float h = 1.0f / (float)NSTEPS;

    for (int L = 0; L < 5; ++L) {
        const float* W1 = oW1 + (size_t)L * H  * H2;
        const float* B1 = ob1 + (size_t)L * H2;
        const float* W2 = oW2 + (size_t)L * H2 * H2;
        const float* B2 = ob2 + (size_t)L * H2;
        const float* W3 = oW3 + (size_t)L * H2 * H;
        const float* B3 = ob3 + (size_t)L * H;

        for (int s = 0; s < NSTEPS; ++s) {
            ode_f(stream, y, k[0], W1, B1, W2, B2, W3, B3, t1, t2);

            combine6<<<cb, 256, 0, stream>>>(ytmp, y, h,
                a21, k[0], 0.f, nullptr, 0.f, nullptr, 0.f, nullptr, 0.f, nullptr, 0.f, nullptr, nState);
            ode_f(stream, ytmp, k[1], W1, B1, W2, B2, W3, B3, t1, t2);

            combine6<<<cb, 256, 0, stream>>>(ytmp, y, h,
                a31, k[0], a32, k[1], 0.f, nullptr, 0.f, nullptr, 0.f, nullptr, 0.f, nullptr, nState);
            ode_f(stream, ytmp, k[2], W1, B1, W2, B2, W3, B3, t1, t2);

            combine6<<<cb, 256, 0, stream>>>(ytmp, y, h,
                a41, k[0], a42, k[1], a43, k[2], 0.f, nullptr, 0.f, nullptr, 0.f, nullptr, nState);
            ode_f(stream, ytmp, k[3], W1, B1, W2, B2, W3, B3, t1, t2);

            combine6<<<cb, 256, 0, stream>>>(ytmp, y, h,
                a51, k[0], a52, k[1], a53, k[2], a54, k[3], 0.f, nullptr, 0.f, nullptr, nState);
            ode_f(stream, ytmp, k[4], W1, B1, W2, B2, W3, B3, t1, t2);

            combine6<<<cb, 256, 0, stream>>>(ytmp, y, h,
                a61, k[0], a62, k[1], a63, k[2], a64, k[3], a65, k[4], 0.f, nullptr, nState);
            ode_f(stream, ytmp, k[5], W1, B1, W2, B2, W3, B3, t1, t2);

            combine6<<<cb, 256, 0, stream>>>(y, y, h,
                b1, k[0], 0.f, nullptr, b3, k[2], b4, k[3], b5, k[4], b6, k[5], nState);
        }
    }

    // ---- output head ----
    launch_gemm(stream, y,  Wo1, bo1, t1,             BATCH, H,    H, 1);
    launch_gemm(stream, t1, Wo2, bo2, (float*)d_out,  BATCH, DOUT, H, 1);
}